// Virtual_GCN_81887846466032
// MI455X (gfx1250) — compile-verified
//
#include <hip/hip_runtime.h>

typedef __attribute__((ext_vector_type(2))) float v2f;
typedef __attribute__((ext_vector_type(8))) float v8f;

#define TT 24      // T
#define NN 4096    // C*C
#define HH 32      // feature width (E=H=O=32)
#define NTILES (NN / 16)

// D = A(16x4) * B(4x16) + C, full f32 (V_WMMA_F32_16X16X4_F32)
__device__ __forceinline__ v8f wmma_f32(v2f a, v2f b, v8f c) {
    return __builtin_amdgcn_wmma_f32_16x16x4_f32(false, a, false, b, (short)0, c, false, false);
}

__global__ __launch_bounds__(1024) void vgcn_kernel(
    const float* __restrict__ Flow,   // (B, 24, 4096)
    const float* __restrict__ Edge,   // (B, 4096, 32)
    const float* __restrict__ W0, const float* __restrict__ b0,
    const float* __restrict__ W1, const float* __restrict__ b1,
    const float* __restrict__ W2, const float* __restrict__ b2,
    float* __restrict__ Out,          // (B, 4096, 32)
    float* __restrict__ ws)           // per batch: xbuf (N*32) + ubuf (N*32)
{
    __shared__ float s_dinv[NN];      // 16 KB
    __shared__ float s_z[32 * 32];    // z = xn @ u, rows 24..31 padded zero
    __shared__ float s_W[32 * 32];
    __shared__ float s_b[32];
    __shared__ float s_rn[32];        // 1/||flow_t||
    __shared__ float s_c[32];         // rn[t]^2 * sum(flow_t)

    const int bidx = blockIdx.x;
    const int tid  = threadIdx.x;
    const int wave = tid >> 5;        // 0..31
    const int lane = tid & 31;
    const int g    = lane >> 4;       // half-wave select
    const int m    = lane & 15;

    const float* flow = Flow + (size_t)bidx * TT * NN;
    const float* edge = Edge + (size_t)bidx * NN * HH;
    float* xbuf = ws + (size_t)bidx * (2 * NN * HH);
    float* ubuf = xbuf + NN * HH;
    float* outb = Out + (size_t)bidx * NN * HH;

    // ---- setup: per-row norm and sum of flow (wave w handles row w) ----
    if (wave < TT) {
        const float* row = flow + wave * NN;
        float s = 0.f, q = 0.f;
        for (int i = lane; i < NN; i += 32) {
            float v = row[i];
            s += v; q += v * v;
        }
        #pragma unroll
        for (int off = 16; off > 0; off >>= 1) {
            s += __shfl_xor(s, off, 32);
            q += __shfl_xor(q, off, 32);
        }
        if (lane == 0) {
            float rn = 1.0f / fmaxf(sqrtf(q), 1e-12f);
            s_rn[wave] = rn;
            s_c[wave]  = rn * (s * rn);
        }
    } else if (lane == 0) {
        s_rn[wave] = 0.f;
        s_c[wave]  = 0.f;
    }
    __syncthreads();

    // ---- dinv[i] = rsqrt(1 + sum_t flow[t,i] * c[t])  (deg>0 guard) ----
    for (int i = tid; i < NN; i += 1024) {
        float deg = 1.0f;
        #pragma unroll
        for (int t = 0; t < TT; ++t) deg += flow[t * NN + i] * s_c[t];
        s_dinv[i] = (deg > 0.f) ? rsqrtf(deg) : 0.f;
    }
    __syncthreads();

    const float* Ws[3] = {W0, W1, W2};
    const float* bs[3] = {b0, b1, b2};

    for (int layer = 0; layer < 3; ++layer) {
        for (int i = tid; i < 32 * 32; i += 1024) { s_W[i] = Ws[layer][i]; s_z[i] = 0.f; }
        if (tid < 32) s_b[tid] = bs[layer][tid];
        __syncthreads();

        const float* x = (layer == 0) ? edge : xbuf;

        // ---- stage 1: u = dinv ⊙ (x @ W), N x 32, K = 32 ----
        for (int tile = wave; tile < NTILES; tile += 32) {
            const int i0 = tile * 16;
            v8f acc0 = {}, acc1 = {};
            #pragma unroll
            for (int k = 0; k < 32; k += 4) {
                v2f a;
                a.x = x[(i0 + m) * HH + k + 2 * g];
                a.y = x[(i0 + m) * HH + k + 2 * g + 1];
                v2f bA, bB;
                bA.x = s_W[(k + 2 * g) * 32 + m];
                bA.y = s_W[(k + 2 * g + 1) * 32 + m];
                bB.x = s_W[(k + 2 * g) * 32 + 16 + m];
                bB.y = s_W[(k + 2 * g + 1) * 32 + 16 + m];
                acc0 = wmma_f32(a, bA, acc0);
                acc1 = wmma_f32(a, bB, acc1);
            }
            #pragma unroll
            for (int r = 0; r < 8; ++r) {
                int row = i0 + r + 8 * g;
                float dv = s_dinv[row];
                ubuf[row * HH + m]      = dv * acc0[r];
                ubuf[row * HH + 16 + m] = dv * acc1[r];
            }
        }
        __syncthreads();

        // ---- stage 2: z = xn @ u (24x32), K=4096 split across 32 waves ----
        {
            v8f za00 = {}, za01 = {}, za10 = {}, za11 = {};
            const int k0 = wave * (NN / 32);
            for (int k = k0; k < k0 + NN / 32; k += 4) {
                float rn0 = s_rn[m];                       // t = m (0..15)
                v2f a0, a1;
                a0.x = flow[m * NN + k + 2 * g] * rn0;
                a0.y = flow[m * NN + k + 2 * g + 1] * rn0;
                int  t1 = 16 + m;                          // t = 16..31 (24+ zero)
                bool ok = (t1 < TT);
                float rn1 = ok ? s_rn[t1] : 0.f;
                a1.x = ok ? flow[t1 * NN + k + 2 * g] * rn1 : 0.f;
                a1.y = ok ? flow[t1 * NN + k + 2 * g + 1] * rn1 : 0.f;
                v2f bA, bB;
                bA.x = ubuf[(k + 2 * g) * HH + m];
                bA.y = ubuf[(k + 2 * g + 1) * HH + m];
                bB.x = ubuf[(k + 2 * g) * HH + 16 + m];
                bB.y = ubuf[(k + 2 * g + 1) * HH + 16 + m];
                za00 = wmma_f32(a0, bA, za00);
                za01 = wmma_f32(a0, bB, za01);
                za10 = wmma_f32(a1, bA, za10);
                za11 = wmma_f32(a1, bB, za11);
            }
            #pragma unroll
            for (int r = 0; r < 8; ++r) {
                int t0 = r + 8 * g;
                atomicAdd(&s_z[t0 * 32 + m],             za00[r]);
                atomicAdd(&s_z[t0 * 32 + 16 + m],        za01[r]);
                atomicAdd(&s_z[(16 + t0) * 32 + m],      za10[r]);
                atomicAdd(&s_z[(16 + t0) * 32 + 16 + m], za11[r]);
            }
        }
        __syncthreads();

        // ---- stage 3: out = relu(dinv ⊙ ((xn^T @ z) + u) + b), K = 24 ----
        float* dst = (layer == 2) ? outb : xbuf;
        for (int tile = wave; tile < NTILES; tile += 32) {
            const int i0 = tile * 16;
            v8f acc0 = {}, acc1 = {};
            #pragma unroll
            for (int k = 0; k < TT; k += 4) {
                v2f a;  // A[m][kk] = xn[k+kk][i0+m] — lane-contiguous loads
                a.x = flow[(k + 2 * g) * NN + i0 + m] * s_rn[k + 2 * g];
                a.y = flow[(k + 2 * g + 1) * NN + i0 + m] * s_rn[k + 2 * g + 1];
                v2f bA, bB;
                bA.x = s_z[(k + 2 * g) * 32 + m];
                bA.y = s_z[(k + 2 * g + 1) * 32 + m];
                bB.x = s_z[(k + 2 * g) * 32 + 16 + m];
                bB.y = s_z[(k + 2 * g + 1) * 32 + 16 + m];
                acc0 = wmma_f32(a, bA, acc0);
                acc1 = wmma_f32(a, bB, acc1);
            }
            #pragma unroll
            for (int r = 0; r < 8; ++r) {
                int row = i0 + r + 8 * g;
                float dv = s_dinv[row];
                float v0 = dv * (acc0[r] + ubuf[row * HH + m])      + s_b[m];
                float v1 = dv * (acc1[r] + ubuf[row * HH + 16 + m]) + s_b[16 + m];
                dst[row * HH + m]      = fmaxf(v0, 0.f);
                dst[row * HH + 16 + m] = fmaxf(v1, 0.f);
            }
        }
        __syncthreads();
    }
}

extern "C" void kernel_launch(void* const* d_in, const int* in_sizes, int n_in,
                              void* d_out, int out_size, void* d_ws, size_t ws_size,
                              hipStream_t stream) {
    const float* Flow = (const float*)d_in[0];
    const float* Edge = (const float*)d_in[1];
    const float* W0   = (const float*)d_in[2];
    const float* b0   = (const float*)d_in[3];
    const float* W1   = (const float*)d_in[4];
    const float* b1   = (const float*)d_in[5];
    const float* W2   = (const float*)d_in[6];
    const float* b2   = (const float*)d_in[7];
    // one workgroup (32 waves) per batch; needs 8 MB of d_ws (2 * N * 32 f32 per batch)
    vgcn_kernel<<<8, 1024, 0, stream>>>(Flow, Edge, W0, b0, W1, b1, W2, b2,
                                        (float*)d_out, (float*)d_ws);
}